// ConstrainedDeformableConv2d3DoF_61151744360470
// MI455X (gfx1250) — compile-verified
//
#include <hip/hip_runtime.h>
#include <math.h>

typedef __attribute__((ext_vector_type(2))) float v2f;
typedef __attribute__((ext_vector_type(8))) float v8f;

#define HH   128
#define WW   128
#define HW   16384        // 128*128
#define CIN  64
#define OCN  64
#define KT   9
#define KDIM 576          // CIN*KT
#define MTILE 32
#define CBLK  8
#define KCH   72          // CBLK*KT
#define ASTR  74          // padded A stride (even -> float2-aligned, spreads banks)
#define BSTR  65          // padded B stride (conflict-free transposed fill)

// ---------------------------------------------------------------------------
// Kernel 1: 3-channel 3x3 conv (pad 1) + bias -> build 18 offset channels per
// pixel, stored per-pixel as [offy0..offy8, offx0..offx8] (reference concat
// order; consumer indexes chan[2k] / chan[2k+1] to reproduce the reshape).
// ---------------------------------------------------------------------------
__global__ __launch_bounds__(256) void offset_build_kernel(
    const float* __restrict__ x, const float* __restrict__ ow,
    const float* __restrict__ ob, float* __restrict__ offbuf) {
  int gpix = blockIdx.x * blockDim.x + threadIdx.x;   // 65536 pixels
  int b  = gpix >> 14;
  int hw = gpix & (HW - 1);
  int ho = hw >> 7;
  int wo = hw & 127;

  float acc0 = ob[0], acc1 = ob[1], acc2 = ob[2];
  const float* xb = x + (size_t)b * CIN * HW;

  for (int cin = 0; cin < CIN; ++cin) {
    const float* xc = xb + (size_t)cin * HW;
    float v[9];
#pragma unroll
    for (int t = 0; t < 9; ++t) {
      int yy = ho + (t / 3) - 1;
      int xx = wo + (t % 3) - 1;
      bool ok = (yy >= 0) & (yy < HH) & (xx >= 0) & (xx < WW);
      v[t] = ok ? xc[yy * WW + xx] : 0.f;
    }
    const float* w0 = ow + cin * 9;       // ow[oc][cin][t], oc stride = 576
#pragma unroll
    for (int t = 0; t < 9; ++t) {
      acc0 = fmaf(v[t], w0[t], acc0);
      acc1 = fmaf(v[t], w0[576 + t], acc1);
      acc2 = fmaf(v[t], w0[1152 + t], acc2);
    }
  }

  float oyp1 = acc0 + 1.f;
  float oxp1 = acc1 + 1.f;
  float s, c;
  __sincosf(acc2, &s, &c);
  float A  = oxp1 * s;   // oxp1*sin
  float Bc = oyp1 * c;   // oyp1*cos
  float Cc = oxp1 * c;   // oxp1*cos
  float D  = oyp1 * s;   // oyp1*sin

  float offy[9] = {-A - Bc + 1.f, -A, -A + Bc - 1.f,
                   -Bc + 1.f,     0.f,  Bc - 1.f,
                    A - Bc + 1.f,  A,   A + Bc - 1.f};
  float offx[9] = {-Cc + D + 1.f, -Cc + 1.f, -Cc - D + 1.f,
                    D,             0.f,      -D,
                    Cc + D - 1.f,  Cc - 1.f,  Cc - D - 1.f};

  float* op = offbuf + (size_t)gpix * 18;
#pragma unroll
  for (int k = 0; k < 9; ++k) { op[k] = offy[k]; op[9 + k] = offx[k]; }
}

// ---------------------------------------------------------------------------
// Kernel 2: fused bilinear gather + GEMM via V_WMMA_F32_16X16X4_F32.
// M = B*Ho*Wo = 65536 pixels, N = 64 out channels, K = 576 (= 64 cin * 9 taps).
// Workgroup: 256 thr (8 waves), 32-pixel M-tile x full 64 N-tile.
// Gather uses paired 8-byte loads: x-adjacent bilinear corners are contiguous;
// border clamping is folded into pre-scattered float2 weight vectors so every
// load stays in-bounds and invalid corners contribute zero.
// ---------------------------------------------------------------------------
__global__ __launch_bounds__(256) void deform_wmma_kernel(
    const float* __restrict__ x, const float* __restrict__ offbuf,
    const float* __restrict__ weight, float* __restrict__ out) {
  __shared__ float As[MTILE * ASTR];      // 32 x 72 (padded)
  __shared__ float Bs[KCH * BSTR];        // 72 x 64 (padded)
  __shared__ int   mi2[MTILE * KT * 2];   // 2 row-pair base indices /(pix,tap)
  __shared__ float mw4[MTILE * KT * 4];   // 4 pre-scattered bilinear weights

  const int tid       = threadIdx.x;
  const int tile_base = blockIdx.x * MTILE;
  const int b         = tile_base >> 14;         // 16384 % 32 == 0
  const float* xb     = x + (size_t)b * CIN * HW;

  // ---- per-tile sampling metadata (32 pixels x 9 taps) ----
  for (int e = tid; e < MTILE * KT; e += 256) {
    int p = e / KT, k = e - p * KT;
    int gpix = tile_base + p;
    int hw = gpix & (HW - 1);
    int ho = hw >> 7, wo = hw & 127;
    const float* op = offbuf + (size_t)gpix * 18;
    // reshape(B,18,..)->(B,9,2,..): tap k y-offset = chan 2k, x-offset = 2k+1
    float oy = op[2 * k];
    float ox = op[2 * k + 1];
    float py = (float)(ho - 1 + k / 3) + oy;
    float px = (float)(wo - 1 + k % 3) + ox;
    float y0f = floorf(py), x0f = floorf(px);
    int   y0 = (int)y0f,    x0 = (int)x0f;
    float wy1 = py - y0f, wy0 = 1.f - wy1;
    float wx1 = px - x0f, wx0 = 1.f - wx1;

    // validity masks per corner (row x col)
    bool r0v = (y0 >= 0) & (y0 < HH);
    bool r1v = (y0 + 1 >= 0) & (y0 + 1 < HH);
    bool c0v = (x0 >= 0) & (x0 < WW);
    bool c1v = (x0 + 1 >= 0) & (x0 + 1 < WW);
    float w00 = (r0v & c0v) ? wy0 * wx0 : 0.f;
    float w01 = (r0v & c1v) ? wy0 * wx1 : 0.f;
    float w10 = (r1v & c0v) ? wy1 * wx0 : 0.f;
    float w11 = (r1v & c1v) ? wy1 * wx1 : 0.f;

    // pair base column and per-corner selector within the loaded pair
    int xbase = x0 < 0 ? 0 : (x0 > WW - 2 ? WW - 2 : x0);
    int xc0 = x0 < 0 ? 0 : (x0 > WW - 1 ? WW - 1 : x0);
    int xc1 = x0 + 1 < 0 ? 0 : (x0 + 1 > WW - 1 ? WW - 1 : x0 + 1);
    int s0 = xc0 - xbase;                  // 0 or 1
    int s1 = xc1 - xbase;                  // 0 or 1
    int y0c = y0 < 0 ? 0 : (y0 > HH - 1 ? HH - 1 : y0);
    int y1c = y0 + 1 < 0 ? 0 : (y0 + 1 > HH - 1 ? HH - 1 : y0 + 1);

    // scatter weights into the two pair slots
    float wv[4] = {0.f, 0.f, 0.f, 0.f};    // {row0[0],row0[1],row1[0],row1[1]}
    wv[s0]     += w00;
    wv[s1]     += w01;
    wv[2 + s0] += w10;
    wv[2 + s1] += w11;

    mi2[e * 2 + 0] = y0c * WW + xbase;
    mi2[e * 2 + 1] = y1c * WW + xbase;
    mw4[e * 4 + 0] = wv[0];
    mw4[e * 4 + 1] = wv[1];
    mw4[e * 4 + 2] = wv[2];
    mw4[e * 4 + 3] = wv[3];
  }
  __syncthreads();

  // ---- wave tiling: 8 waves -> 2 m-rows x 4 n-cols of 16x16 tiles ----
  const int lane = tid & 31;
  const int wave = tid >> 5;
  const int mrow = (wave >> 2) * 16;   // 0 or 16 (pixel offset in tile)
  const int ncol = (wave & 3) * 16;    // 0,16,32,48 (out-channel offset)
  const int half = lane >> 4;          // K half (lanes 16-31 hold K=2,3)
  const int l16  = lane & 15;

  v8f acc = {};

  for (int cb = 0; cb < CIN / CBLK; ++cb) {
    // Stage B chunk: Bs[r][n] = weight[n][cb*72 + r]   (coalesced global read)
    for (int e = tid; e < KCH * OCN; e += 256) {
      int n = e / KCH, rr = e - n * KCH;
      Bs[rr * BSTR + n] = weight[(size_t)n * KDIM + cb * KCH + rr];
    }
    // Stage A chunk: As[p][r] = bilinear sample of x[b, cb*8 + r/9, tap r%9]
    // Two 8-byte pair loads per sample instead of four scalar loads.
    for (int e = tid; e < MTILE * KCH; e += 256) {
      int p = e / KCH, r = e - p * KCH;
      int cl = r / KT, k = r - cl * KT;
      int me = p * KT + k;
      const float* xc = xb + (size_t)(cb * CBLK + cl) * HW;
      float pr0[2], pr1[2];
      __builtin_memcpy(pr0, xc + mi2[me * 2 + 0], 8);
      __builtin_memcpy(pr1, xc + mi2[me * 2 + 1], 8);
      float v = pr0[0] * mw4[me * 4 + 0] + pr0[1] * mw4[me * 4 + 1] +
                pr1[0] * mw4[me * 4 + 2] + pr1[1] * mw4[me * 4 + 3];
      As[p * ASTR + r] = v;
    }
    __syncthreads();

    // 18 back-to-back fp32 WMMAs over the 72-deep chunk
#pragma unroll
    for (int rr = 0; rr < KCH; rr += 4) {
      // A 16x4 frag: lane half selects K pair {0,1} or {2,3}; one ds_load_b64
      v2f a = *(const v2f*)&As[(mrow + l16) * ASTR + rr + 2 * half];
      v2f bf;
      bf.x = Bs[(rr + 2 * half) * BSTR + ncol + l16];
      bf.y = Bs[(rr + 2 * half + 1) * BSTR + ncol + l16];
      acc = __builtin_amdgcn_wmma_f32_16x16x4_f32(
          /*neg_a=*/false, a, /*neg_b=*/false, bf,
          /*c_mod=*/(short)0, acc, /*reuse_a=*/false, /*reuse_b=*/false);
    }
    __syncthreads();
  }

  // ---- epilogue: C/D layout = VGPR j -> M=j (lanes 0-15) / M=j+8 (16-31) ----
#pragma unroll
  for (int j = 0; j < 8; ++j) {
    int m = mrow + j + 8 * half;
    int n = ncol + l16;
    int gpix = tile_base + m;
    int hw = gpix & (HW - 1);
    out[((size_t)b * OCN + n) * HW + hw] = acc[j];
  }
}

// ---------------------------------------------------------------------------
extern "C" void kernel_launch(void* const* d_in, const int* in_sizes, int n_in,
                              void* d_out, int out_size, void* d_ws, size_t ws_size,
                              hipStream_t stream) {
  const float* x      = (const float*)d_in[0];   // (4,64,128,128)
  const float* ow     = (const float*)d_in[1];   // (3,64,3,3)
  const float* ob     = (const float*)d_in[2];   // (3,)
  const float* weight = (const float*)d_in[3];   // (64,64,3,3)
  float* out    = (float*)d_out;                 // (4,64,128,128)
  float* offbuf = (float*)d_ws;                  // 65536*18 floats = 4.5 MB

  offset_build_kernel<<<(4 * HW) / 256, 256, 0, stream>>>(x, ow, ob, offbuf);
  deform_wmma_kernel<<<(4 * HW) / MTILE, 256, 0, stream>>>(x, offbuf, weight, out);
}